// DBi_LSTM_88227218194656
// MI455X (gfx1250) — compile-verified
//
#include <hip/hip_runtime.h>

// ---------------------------------------------------------------------------
// DBi-LSTM (5 bidirectional layers, H=45, B=128, T=512) + FC(90->128)+ReLU
// (1) parallel WMMA GEMM precomputes x@Wih^T + b for all (b,t) per layer;
// (2) persistent recurrent kernel (8 batch-tiles x 2 dirs) runs the 512-step
//     scan with h/c/Whh^T resident on-chip, fp32 WMMA for h@Whh^T, prefetch
//     of next-step gates; (3) WMMA FC + ReLU epilogue.
// ---------------------------------------------------------------------------

typedef float v2f __attribute__((ext_vector_type(2)));
typedef float v8f __attribute__((ext_vector_type(8)));

#define Hh    45          // hidden size
#define G4    180         // 4*H gates
#define NP    192         // gates padded to 12 WMMA col-tiles
#define Bb    128         // batch
#define Tt    512         // time steps
#define HSTR  50          // LDS row stride for h buffer (floats, conflict-free)
#define PSTR  (2 * NP)    // LDS row stride for pair-interleaved Whh^T

__device__ __forceinline__ float fastrcp(float x) {
    return __builtin_amdgcn_rcpf(x);              // v_rcp_f32 (1 ulp)
}
__device__ __forceinline__ float sigf(float x) {
    return fastrcp(1.0f + __expf(-x));
}
__device__ __forceinline__ float tanhf_(float x) {
    float e = __expf(fminf(fmaxf(2.0f * x, -30.0f), 30.0f));
    return (e - 1.0f) * fastrcp(e + 1.0f);
}
__device__ __forceinline__ int iclamp(int a, int hi) { return a < hi ? a : hi; }

// D = A(16x4) * B(4x16) + C, fp32 WMMA
__device__ __forceinline__ v8f wmma4(v2f a, v2f b, v8f c) {
    return __builtin_amdgcn_wmma_f32_16x16x4_f32(false, a, false, b,
                                                 (short)0, c, false, false);
}

// ---------------------------------------------------------------------------
// Kernel 1: gate precompute  Gout[dir][b][t][n] = X[b][t][:] @ Wih[dir][n][:]^T
//                                                 + bih[dir][n] + bhh[dir][n]
// M = b*T + t (65536 rows). CONTIG=true when X k-stride == 1 (layers 1..4).
// Branch-free: full-K main loop + one clamped/selected tail iteration.
// Padded gate columns (n>=180) receive garbage; never read downstream.
// ---------------------------------------------------------------------------
template <bool CONTIG>
__global__ void gate_precompute(const float* __restrict__ X,
                                long sb, long st, long sk, int Kin,
                                const float* __restrict__ Wih,   // [2][G4][Kin]
                                const float* __restrict__ bih,   // [2][G4]
                                const float* __restrict__ bhh,   // [2][G4]
                                float* __restrict__ Gout)        // [2][Bb][Tt][NP]
{
    const int tid   = threadIdx.x;
    const int lane  = tid & 31;
    const int wave  = tid >> 5;                     // 8 waves / block
    const int mrow0 = (blockIdx.x * 8 + wave) * 16; // M-tile base (M = b*Tt + t)
    const int n0    = blockIdx.y * 16;
    const int dir   = blockIdx.z;
    const int col   = lane & 15;
    const int rhi   = (lane >> 4) * 8;              // C/D row half
    const int kpair = (lane >> 4) * 2;              // A/B k pair

    const float* wih = Wih + (size_t)dir * G4 * Kin;
    const int Mrow = mrow0 + col;                   // A-fragment row
    const int bA = Mrow >> 9;                       // /Tt
    const int tA = Mrow & (Tt - 1);
    const float* xrow = X + (size_t)bA * sb + (size_t)tA * st;
    const int nc = iclamp(n0 + col, G4 - 1);        // clamped weight row
    const float* wrow = wih + (size_t)nc * Kin;

    v8f acc = {};
    int k0 = 0;
    // main loop: every ka in [k0, k0+3] is in range -> unconditional loads
    for (; k0 + 4 <= Kin; k0 += 4) {
        const int ka = k0 + kpair;
        v2f a, b;
        if (CONTIG) {
            a = *(const v2f*)(xrow + ka);
        } else {
            a.x = xrow[(size_t)ka * sk];
            a.y = xrow[(size_t)(ka + 1) * sk];
        }
        b = *(const v2f*)(wrow + ka);
        acc = wmma4(a, b, acc);
    }
    // tail: clamp addresses (always legal), select values (v_cndmask)
    const int Kp = (Kin + 3) & ~3;
    for (; k0 < Kp; k0 += 4) {
        const int ka = k0 + kpair;
        const int kc0 = iclamp(ka, Kin - 1);
        const int kc1 = iclamp(ka + 1, Kin - 1);
        v2f a, b;
        float ax, ay;
        if (CONTIG) { ax = xrow[kc0]; ay = xrow[kc1]; }
        else        { ax = xrow[(size_t)kc0 * sk]; ay = xrow[(size_t)kc1 * sk]; }
        a.x = (ka     < Kin) ? ax : 0.0f;
        a.y = (ka + 1 < Kin) ? ay : 0.0f;
        b.x = (ka     < Kin) ? wrow[kc0] : 0.0f;
        b.y = (ka + 1 < Kin) ? wrow[kc1] : 0.0f;
        acc = wmma4(a, b, acc);
    }

    const float bias = bih[dir * G4 + nc] + bhh[dir * G4 + nc];
    const int n = n0 + col;
#pragma unroll
    for (int r = 0; r < 8; ++r) {
        const int Mr = mrow0 + rhi + r;
        const int b_ = Mr >> 9;
        const int t_ = Mr & (Tt - 1);
        Gout[((((size_t)dir * Bb + b_) * Tt) + t_) * NP + n] = acc[r] + bias;
    }
}

// ---------------------------------------------------------------------------
// Kernel 2: persistent recurrent scan. grid = (8 batch tiles, 2 directions),
// 384 threads = 12 waves = 12 N-tiles of one 16-row batch tile. Per step:
// prefetch next-step gates; acc <- precomputed gates; acc += h_prev @ Whh^T
// (12x wmma_f32_16x16x4, A and B fragments as single ds_load_b64); spill
// gates to LDS; elementwise LSTM cell (c in registers); h -> LDS + out.
// ---------------------------------------------------------------------------
__global__ void lstm_recurrent(const float* __restrict__ G,    // [2][Bb][Tt][NP]
                               const float* __restrict__ Whh,  // [2][G4][Hh]
                               float* __restrict__ Y)          // [Bb][Tt][2*Hh]
{
    const int tid   = threadIdx.x;
    const int lane  = tid & 31;
    const int wave  = tid >> 5;            // 0..11 -> N tile
    const int dir   = blockIdx.y;
    const int mb0   = blockIdx.x * 16;     // batch rows [mb0, mb0+16)
    const int col   = lane & 15;
    const int rhi   = (lane >> 4) * 8;
    const int kpair = (lane >> 4) * 2;
    const int n0    = wave * 16;

    extern __shared__ float smem[];
    float* whhT2 = smem;                   // [24][NP][2] pair-interleaved Whh^T
    float* hbuf  = smem + 24 * PSTR;       // [16][HSTR]
    float* gbuf  = hbuf + 16 * HSTR;       // [16][NP]

    // Stage Whh^T pair-interleaved: whhT2[k/2][n][k&1] = Whh[n][k], 0-padded
    const float* whh = Whh + (size_t)dir * G4 * Hh;
    for (int i = tid; i < 48 * NP; i += blockDim.x) {
        const int k = i / NP, nn = i % NP;
        whhT2[(k >> 1) * PSTR + nn * 2 + (k & 1)] =
            (k < Hh && nn < G4) ? whh[nn * Hh + k] : 0.0f;
    }
    for (int i = tid; i < 16 * HSTR; i += blockDim.x) hbuf[i] = 0.0f;
    float c0 = 0.0f, c1 = 0.0f;
    __syncthreads();

    // hoisted per-thread elementwise indices / pointers
    const int bl0 = tid / Hh, u0 = tid - bl0 * Hh;        // tid < 720 always
    const int idx1 = tid + 384;
    const bool act1 = idx1 < 16 * Hh;
    const int bl1 = idx1 / Hh, u1 = idx1 - bl1 * Hh;
    float* y0 = Y + ((size_t)(mb0 + bl0) * Tt) * (2 * Hh) + dir * Hh + u0;
    float* y1 = Y + ((size_t)(mb0 + bl1) * Tt) * (2 * Hh) + dir * Hh + u1;
    // per-wave gate base: row (mb0+row), col (n0+col), at time t
    const float* Gw  = G + (((size_t)dir * Bb + mb0) * Tt) * NP + n0 + col;
    const float* Gpf = G + (((size_t)dir * Bb + mb0 + col) * Tt) * NP + n0;

    for (int s = 0; s < Tt; ++s) {
        const int t = dir ? (Tt - 1 - s) : s;
        // prefetch next step's gate slice (global_prefetch_b8) to hide latency
        const int tn = dir ? (s + 2 <= Tt ? Tt - 2 - s : 0)
                           : (s + 1 < Tt ? s + 1 : Tt - 1);
        __builtin_prefetch(Gpf + (size_t)tn * NP, 0, 0);

        // accumulator init = precomputed input gates (x@Wih^T + b)
        v8f acc;
#pragma unroll
        for (int r = 0; r < 8; ++r)
            acc[r] = Gw[((size_t)(rhi + r) * Tt + t) * NP];

        // acc += h_prev @ Whh^T   (K = 48, twelve K=4 WMMAs)
#pragma unroll
        for (int k0 = 0; k0 < 48; k0 += 4) {
            const v2f a = *(const v2f*)&hbuf[col * HSTR + k0 + kpair];
            const v2f b = *(const v2f*)&whhT2[((k0 + kpair) >> 1) * PSTR
                                              + (n0 + col) * 2];
            acc = wmma4(a, b, acc);
        }
        // spill gates
#pragma unroll
        for (int r = 0; r < 8; ++r)
            gbuf[(rhi + r) * NP + n0 + col] = acc[r];
        __syncthreads();

        // elementwise LSTM cell: 16*45 = 720 units over 384 threads
        {
            const float gi = gbuf[bl0 * NP + u0];
            const float gf = gbuf[bl0 * NP + Hh + u0];
            const float gg = gbuf[bl0 * NP + 2 * Hh + u0];
            const float go = gbuf[bl0 * NP + 3 * Hh + u0];
            const float cn = sigf(gf) * c0 + sigf(gi) * tanhf_(gg);
            const float hn = sigf(go) * tanhf_(cn);
            c0 = cn;
            hbuf[bl0 * HSTR + u0] = hn;
            y0[(size_t)t * (2 * Hh)] = hn;
        }
        if (act1) {
            const float gi = gbuf[bl1 * NP + u1];
            const float gf = gbuf[bl1 * NP + Hh + u1];
            const float gg = gbuf[bl1 * NP + 2 * Hh + u1];
            const float go = gbuf[bl1 * NP + 3 * Hh + u1];
            const float cn = sigf(gf) * c1 + sigf(gi) * tanhf_(gg);
            const float hn = sigf(go) * tanhf_(cn);
            c1 = cn;
            hbuf[bl1 * HSTR + u1] = hn;
            y1[(size_t)t * (2 * Hh)] = hn;
        }
        __syncthreads();
    }
}

// ---------------------------------------------------------------------------
// Kernel 3: out = relu( h_last(128x90) @ fcW^T(90x128) + fcb ), WMMA tiles.
// ---------------------------------------------------------------------------
__global__ void fc_relu(const float* __restrict__ Hl,   // [Bb][Tt][2*Hh]
                        const float* __restrict__ fcW,  // [128][90]
                        const float* __restrict__ fcb,  // [128]
                        float* __restrict__ out)        // [128][128]
{
    const int tid   = threadIdx.x;
    const int lane  = tid & 31;
    const int wave  = tid >> 5;          // 8 waves -> N tiles
    const int m0    = blockIdx.x * 16;
    const int n0    = wave * 16;
    const int col   = lane & 15;
    const int rhi   = (lane >> 4) * 8;
    const int kpair = (lane >> 4) * 2;
    const int K = 2 * Hh;                // 90

    const float* arow = Hl + ((size_t)(m0 + col) * Tt + (Tt - 1)) * K;
    const float* brow = fcW + (size_t)(n0 + col) * K;
    v8f acc = {};
    int k0 = 0;
    for (; k0 + 4 <= K; k0 += 4) {       // full iterations, unconditional b64
        const int ka = k0 + kpair;
        const v2f a = *(const v2f*)(arow + ka);
        const v2f b = *(const v2f*)(brow + ka);
        acc = wmma4(a, b, acc);
    }
    {                                    // tail (ka in {88,89} valid, {90,91} not)
        const int ka = k0 + kpair;
        const int kc0 = iclamp(ka, K - 1), kc1 = iclamp(ka + 1, K - 1);
        v2f a, b;
        a.x = (ka     < K) ? arow[kc0] : 0.0f;
        a.y = (ka + 1 < K) ? arow[kc1] : 0.0f;
        b.x = (ka     < K) ? brow[kc0] : 0.0f;
        b.y = (ka + 1 < K) ? brow[kc1] : 0.0f;
        acc = wmma4(a, b, acc);
    }
    const float bias = fcb[n0 + col];
#pragma unroll
    for (int r = 0; r < 8; ++r)
        out[(size_t)(m0 + rhi + r) * 128 + n0 + col] = fmaxf(acc[r] + bias, 0.0f);
}

// ---------------------------------------------------------------------------
extern "C" void kernel_launch(void* const* d_in, const int* in_sizes, int n_in,
                              void* d_out, int out_size, void* d_ws, size_t ws_size,
                              hipStream_t stream) {
    (void)in_sizes; (void)n_in; (void)out_size; (void)ws_size;
    const float* x      = (const float*)d_in[0];   // (128, 3, 512)
    const float* Wih_l0 = (const float*)d_in[1];   // (2, 180, 3)
    const float* Whh_l0 = (const float*)d_in[2];   // (2, 180, 45)
    const float* bih_l0 = (const float*)d_in[3];   // (2, 180)
    const float* bhh_l0 = (const float*)d_in[4];   // (2, 180)
    const float* Wih_r  = (const float*)d_in[5];   // (4, 2, 180, 90)
    const float* Whh_r  = (const float*)d_in[6];   // (4, 2, 180, 45)
    const float* bih_r  = (const float*)d_in[7];   // (4, 2, 180)
    const float* bhh_r  = (const float*)d_in[8];   // (4, 2, 180)
    const float* fcW    = (const float*)d_in[9];   // (128, 90)
    const float* fcb    = (const float*)d_in[10];  // (128,)
    float* out = (float*)d_out;

    char* ws = (char*)d_ws;
    float* G    = (float*)ws;                                   // 2*128*512*192 f32
    float* bufA = (float*)(ws + (size_t)2 * Bb * Tt * NP * 4);  // 128*512*90 f32
    float* bufB = bufA + (size_t)Bb * Tt * (2 * Hh);

    const dim3 gpre(512, 12, 2), bpre(256);
    const dim3 grec(8, 2),       brec(384);
    const size_t smem = (24 * PSTR + 16 * HSTR + 16 * NP) * sizeof(float);

    // Layer 0: read x (B,3,T) directly: stride(b)=1536, stride(t)=1, stride(k)=512
    gate_precompute<false><<<gpre, bpre, 0, stream>>>(x, 1536, 1, 512, 3,
                                                      Wih_l0, bih_l0, bhh_l0, G);
    lstm_recurrent<<<grec, brec, smem, stream>>>(G, Whh_l0, bufA);

    float* cur = bufA;
    float* nxt = bufB;
    for (int l = 0; l < 4; ++l) {
        gate_precompute<true><<<gpre, bpre, 0, stream>>>(
            cur, (long)Tt * (2 * Hh), (long)(2 * Hh), 1L, 2 * Hh,
            Wih_r + (size_t)l * 2 * G4 * (2 * Hh),
            bih_r + (size_t)l * 2 * G4,
            bhh_r + (size_t)l * 2 * G4, G);
        lstm_recurrent<<<grec, brec, smem, stream>>>(
            G, Whh_r + (size_t)l * 2 * G4 * Hh, nxt);
        float* tmp = cur; cur = nxt; nxt = tmp;
    }
    fc_relu<<<dim3(8), dim3(256), 0, stream>>>(cur, fcW, fcb, out);
}